// EGRETLayer_84542136254807
// MI455X (gfx1250) — compile-verified
//
#include <hip/hip_runtime.h>
#include <math.h>

typedef float v2f __attribute__((ext_vector_type(2)));
typedef float v8f __attribute__((ext_vector_type(8)));

#define NN 50000
#define NE 1600000
#define IN_F 64
#define OUT_F 32
#define ED_F 8
#define N_TILES (NN / 16)           // 3125, exact
#define NEG_INF_KEY 0x007FFFFFu     // fkey(-inf)

__device__ __forceinline__ unsigned fkey(float f) {
    unsigned b = __float_as_uint(f);
    return (b & 0x80000000u) ? ~b : (b | 0x80000000u);
}
__device__ __forceinline__ float funkey(unsigned k) {
    return __uint_as_float((k & 0x80000000u) ? (k ^ 0x80000000u) : ~k);
}

// ---------------- Kernel 1: z = x @ W_fc^T + b_fc  (WMMA f32 16x16x4) ----
__global__ void node_fc_kernel(const float* __restrict__ x,
                               const float* __restrict__ W_fc,
                               const float* __restrict__ b_fc,
                               float* __restrict__ z) {
    const int lane = threadIdx.x & 31;
    const int tile = blockIdx.x * (blockDim.x >> 5) + (threadIdx.x >> 5);
    if (tile >= N_TILES) return;
    const int m0   = tile * 16;
    const int half = lane >> 4;     // lanes 16-31 hold K+2 (A) / rows M+8 (C)
    const int l    = lane & 15;
    const int koff = 2 * half;

    v8f c0 = {}; v8f c1 = {};
    const float* arow  = x    + (m0 + l) * IN_F + koff;  // A: row m0+l, K pair
    const float* b0row = W_fc + l        * IN_F + koff;  // B[k][n]=W_fc[n][k], n=l
    const float* b1row = W_fc + (16 + l) * IN_F + koff;  // n = 16+l
#pragma unroll
    for (int t = 0; t < IN_F / 4; ++t) {
        v2f a  = *(const v2f*)(arow  + 4 * t);
        v2f b0 = *(const v2f*)(b0row + 4 * t);
        v2f b1 = *(const v2f*)(b1row + 4 * t);
        c0 = __builtin_amdgcn_wmma_f32_16x16x4_f32(false, a, false, b0, (short)0, c0, false, false);
        c1 = __builtin_amdgcn_wmma_f32_16x16x4_f32(false, a, false, b1, (short)0, c1, false, false);
    }
    const float bias0 = b_fc[l];
    const float bias1 = b_fc[16 + l];
#pragma unroll
    for (int r = 0; r < 8; ++r) {
        const int row = m0 + r + 8 * half;   // C layout: vgpr r -> M=r / r+8
        z[row * OUT_F + l]      = c0[r] + bias0;
        z[row * OUT_F + 16 + l] = c1[r] + bias1;
    }
}

// ---------------- Kernel 2: q = z@W_q^T+b_q, s = z@W_s^T+b_s (WMMA) ------
__global__ void node_qs_kernel(const float* __restrict__ z,
                               const float* __restrict__ W_q, const float* __restrict__ b_q,
                               const float* __restrict__ W_s, const float* __restrict__ b_s,
                               float* __restrict__ qo, float* __restrict__ so) {
    const int lane = threadIdx.x & 31;
    const int tile = blockIdx.x * (blockDim.x >> 5) + (threadIdx.x >> 5);
    if (tile >= N_TILES) return;
    const int m0   = tile * 16;
    const int half = lane >> 4;
    const int l    = lane & 15;
    const int koff = 2 * half;

    v8f q0 = {}, q1 = {}, s0 = {}, s1 = {};
    const float* arow = z   + (m0 + l) * OUT_F + koff;
    const float* bq0  = W_q + l        * OUT_F + koff;
    const float* bq1  = W_q + (16 + l) * OUT_F + koff;
    const float* bs0  = W_s + l        * OUT_F + koff;
    const float* bs1  = W_s + (16 + l) * OUT_F + koff;
#pragma unroll
    for (int t = 0; t < OUT_F / 4; ++t) {
        v2f a = *(const v2f*)(arow + 4 * t);
        q0 = __builtin_amdgcn_wmma_f32_16x16x4_f32(false, a, false, *(const v2f*)(bq0 + 4 * t), (short)0, q0, false, false);
        q1 = __builtin_amdgcn_wmma_f32_16x16x4_f32(false, a, false, *(const v2f*)(bq1 + 4 * t), (short)0, q1, false, false);
        s0 = __builtin_amdgcn_wmma_f32_16x16x4_f32(false, a, false, *(const v2f*)(bs0 + 4 * t), (short)0, s0, false, false);
        s1 = __builtin_amdgcn_wmma_f32_16x16x4_f32(false, a, false, *(const v2f*)(bs1 + 4 * t), (short)0, s1, false, false);
    }
    const float bq_0 = b_q[l], bq_1 = b_q[16 + l];
    const float bs_0 = b_s[l], bs_1 = b_s[16 + l];
#pragma unroll
    for (int r = 0; r < 8; ++r) {
        const int row = m0 + r + 8 * half;
        qo[row * OUT_F + l]      = q0[r] + bq_0;
        qo[row * OUT_F + 16 + l] = q1[r] + bq_1;
        so[row * OUT_F + l]      = s0[r] + bs_0;
        so[row * OUT_F + 16 + l] = s1[r] + bs_1;
    }
}

// ---------------- Kernel 3: init accumulators ----------------------------
__global__ void init_kernel(float* __restrict__ hacc,      // hatt|h2|hez: 3*NN*OUT_F
                            unsigned* __restrict__ maxE, unsigned* __restrict__ maxS,
                            float* __restrict__ sumE, float* __restrict__ sumS) {
    const int i = blockIdx.x * blockDim.x + threadIdx.x;
    if (i < 3 * NN * OUT_F) hacc[i] = 0.0f;
    if (i < NN) {
        maxE[i] = NEG_INF_KEY; maxS[i] = NEG_INF_KEY;
        sumE[i] = 0.0f;        sumS[i] = 0.0f;
    }
}

// ---------------- Kernel 4: per-edge logits + segment max ----------------
__global__ void edge_logits_kernel(const float* __restrict__ z,
                                   const float* __restrict__ qv, const float* __restrict__ sv,
                                   const float* __restrict__ ex,
                                   const int* __restrict__ src, const int* __restrict__ dst,
                                   const float* __restrict__ W_att, const float* __restrict__ b_att,
                                   const float* __restrict__ W_eatt, const float* __restrict__ b_eatt,
                                   float* __restrict__ eL, float* __restrict__ esL,
                                   unsigned* __restrict__ maxE, unsigned* __restrict__ maxS) {
    const int lane = threadIdx.x & 31;
    const int e    = blockIdx.x * (blockDim.x >> 5) + (threadIdx.x >> 5);
    if (e >= NE) return;
    const int s = src[e], d = dst[e];

    const float zs = z[s * OUT_F + lane];
    const float zd = z[d * OUT_F + lane];
    float p = W_att[lane] * zs + W_att[OUT_F + lane] * zd;
    if (lane < ED_F) {
        float ext = b_eatt[lane];
#pragma unroll
        for (int k = 0; k < ED_F; ++k)
            ext += W_eatt[lane * ED_F + k] * ex[e * ED_F + k];
        p += W_att[2 * OUT_F + lane] * ext;
    }
    float pes = qv[s * OUT_F + lane] * sv[d * OUT_F + lane];
#pragma unroll
    for (int off = 16; off > 0; off >>= 1) {
        p   += __shfl_xor(p,   off, 32);
        pes += __shfl_xor(pes, off, 32);
    }
    if (lane == 0) {
        float a  = p + b_att[0];
        float ev = (a >= 0.0f) ? a : 0.2f * a;   // leaky_relu(0.2)
        eL[e]  = ev;
        esL[e] = pes;
        atomicMax(&maxE[d], fkey(ev));
        atomicMax(&maxS[d], fkey(pes));
    }
}

// ---------------- Kernel 5: exp + segment sum ----------------------------
__global__ void edge_exp_kernel(const int* __restrict__ dst,
                                float* __restrict__ eL, float* __restrict__ esL,
                                const unsigned* __restrict__ maxE, const unsigned* __restrict__ maxS,
                                float* __restrict__ sumE, float* __restrict__ sumS) {
    const int e = blockIdx.x * blockDim.x + threadIdx.x;
    if (e >= NE) return;
    const int d = dst[e];
    const float ee = expf(eL[e]  - funkey(maxE[d]));
    const float es = expf(esL[e] - funkey(maxS[d]));
    eL[e] = ee; esL[e] = es;
    atomicAdd(&sumE[d], ee);
    atomicAdd(&sumS[d], es);
}

// ---------------- Kernel 6: alpha/similar + scatter-add ------------------
__global__ void edge_scatter_kernel(const float* __restrict__ z,
                                    const float* __restrict__ ex,
                                    const int* __restrict__ src, const int* __restrict__ dst,
                                    const float* __restrict__ W_edge, const float* __restrict__ b_edge,
                                    const float* __restrict__ expE, const float* __restrict__ expS,
                                    const float* __restrict__ sumE, const float* __restrict__ sumS,
                                    float* __restrict__ hatt, float* __restrict__ h2,
                                    float* __restrict__ hez, float* __restrict__ outAlpha) {
    const int lane = threadIdx.x & 31;
    const int e    = blockIdx.x * (blockDim.x >> 5) + (threadIdx.x >> 5);
    if (e >= NE) return;
    const int s = src[e], d = dst[e];
    const float alpha   = expE[e] / sumE[d];
    const float similar = expS[e] / sumS[d];
    if (lane == 0) outAlpha[e] = alpha;

    const float zs = z[s * OUT_F + lane];
    atomicAdd(&hatt[d * OUT_F + lane], alpha   * zs);
    atomicAdd(&h2  [d * OUT_F + lane], similar * zs);

    float ez = b_edge[lane];
#pragma unroll
    for (int k = 0; k < ED_F; ++k)
        ez += W_edge[lane * ED_F + k] * ex[e * ED_F + k];
    atomicAdd(&hez[d * OUT_F + lane], alpha * ez);
}

// ---------------- Kernel 7: h = hatt*h2 + hez ----------------------------
__global__ void finalize_kernel(const float* __restrict__ hatt, const float* __restrict__ h2,
                                const float* __restrict__ hez, float* __restrict__ out) {
    const int i = blockIdx.x * blockDim.x + threadIdx.x;
    if (i < NN * OUT_F) out[i] = hatt[i] * h2[i] + hez[i];
}

// ---------------- Launch -------------------------------------------------
extern "C" void kernel_launch(void* const* d_in, const int* in_sizes, int n_in,
                              void* d_out, int out_size, void* d_ws, size_t ws_size,
                              hipStream_t stream) {
    const float* x      = (const float*)d_in[0];
    const float* ex     = (const float*)d_in[1];
    const int*   src    = (const int*)  d_in[2];
    const int*   dst    = (const int*)  d_in[3];
    const float* W_fc   = (const float*)d_in[4];
    const float* b_fc   = (const float*)d_in[5];
    const float* W_att  = (const float*)d_in[6];
    const float* b_att  = (const float*)d_in[7];
    const float* W_edge = (const float*)d_in[8];
    const float* b_edge = (const float*)d_in[9];
    const float* W_eatt = (const float*)d_in[10];
    const float* b_eatt = (const float*)d_in[11];
    const float* W_q    = (const float*)d_in[12];
    const float* b_q    = (const float*)d_in[13];
    const float* W_s    = (const float*)d_in[14];
    const float* b_s    = (const float*)d_in[15];

    // workspace layout (floats): z|q|s|hatt|h2|hez|maxE|maxS|sumE|sumS|eL|esL
    float* z    = (float*)d_ws;
    float* q    = z    + NN * OUT_F;
    float* s    = q    + NN * OUT_F;
    float* hatt = s    + NN * OUT_F;
    float* h2   = hatt + NN * OUT_F;
    float* hez  = h2   + NN * OUT_F;
    unsigned* maxE = (unsigned*)(hez + NN * OUT_F);
    unsigned* maxS = maxE + NN;
    float* sumE = (float*)(maxS + NN);
    float* sumS = sumE + NN;
    float* eL   = sumS + NN;
    float* esL  = eL + NE;

    float* out_h     = (float*)d_out;             // [NN, OUT_F]
    float* out_alpha = out_h + NN * OUT_F;        // [NE]

    const int BLK = 256;                          // 8 waves (wave32)
    const int wavesPerBlk = BLK / 32;

    // 1) node fc (WMMA)
    node_fc_kernel<<<(N_TILES + wavesPerBlk - 1) / wavesPerBlk, BLK, 0, stream>>>(x, W_fc, b_fc, z);
    // 2) q/s projections (WMMA)
    node_qs_kernel<<<(N_TILES + wavesPerBlk - 1) / wavesPerBlk, BLK, 0, stream>>>(z, W_q, b_q, W_s, b_s, q, s);
    // 3) init accumulators
    init_kernel<<<(3 * NN * OUT_F + BLK - 1) / BLK, BLK, 0, stream>>>(hatt, maxE, maxS, sumE, sumS);
    // 4) per-edge logits + segment max
    edge_logits_kernel<<<(NE + wavesPerBlk - 1) / wavesPerBlk, BLK, 0, stream>>>(
        z, q, s, ex, src, dst, W_att, b_att, W_eatt, b_eatt, eL, esL, maxE, maxS);
    // 5) exp + segment sums
    edge_exp_kernel<<<(NE + BLK - 1) / BLK, BLK, 0, stream>>>(dst, eL, esL, maxE, maxS, sumE, sumS);
    // 6) alpha/similar + scatter-add
    edge_scatter_kernel<<<(NE + wavesPerBlk - 1) / wavesPerBlk, BLK, 0, stream>>>(
        z, ex, src, dst, W_edge, b_edge, eL, esL, sumE, sumS, hatt, h2, hez, out_alpha);
    // 7) finalize h
    finalize_kernel<<<(NN * OUT_F + BLK - 1) / BLK, BLK, 0, stream>>>(hatt, h2, hez, out_h);
}